// Sampler_32736240730902
// MI455X (gfx1250) — compile-verified
//
#include <hip/hip_runtime.h>
#include <math.h>
#include <stdint.h>

#ifndef __has_builtin
#define __has_builtin(x) 0
#endif

#if __has_builtin(__builtin_amdgcn_tensor_load_to_lds) && \
    __has_builtin(__builtin_amdgcn_s_wait_tensorcnt)
#define HAVE_TDM 1
#else
#define HAVE_TDM 0
#endif

#if __has_builtin(__builtin_amdgcn_global_load_async_to_lds_b32)
#define HAVE_ASYNC_LDS 1
#else
#define HAVE_ASYNC_LDS 0
#endif

#if __has_builtin(__builtin_amdgcn_s_wait_asynccnt)
#define HAVE_WAIT_ASYNC 1
#else
#define HAVE_WAIT_ASYNC 0
#endif

namespace {

constexpr int kV       = 50257;   // vocab size (reference constant)
constexpr int kVPad    = 50264;   // padded float count (16B multiple)
constexpr int kNT      = 256;     // threads per block (8 wave32)
constexpr int kNB1     = 4096;    // 12-bit radix bins
constexpr int kHistMax = 256;     // max history tokens held in LDS
constexpr int kMaskW   = (kV + 31) / 32;  // 1571 words vocab bitmask

struct Scal {
  float    M;        // row max of z
  float    pS;       // top_p * S_total
  float    totalP;   // running total mass
  unsigned totalC;
  int      minT;     // arbitration for cutoff owner
  unsigned found;
  unsigned bin;      // cutoff bin
  unsigned Cb;       // count strictly above cutoff bin
  float    Pb;       // mass strictly above cutoff bin
  unsigned cntb;     // count inside cutoff bin
  float    massb;    // mass inside cutoff bin
};

// Dynamic LDS layout (bytes)
constexpr size_t OFF_Z    = 0;
constexpr size_t OFF_HC   = OFF_Z    + sizeof(float)    * kVPad;    // hist counts
constexpr size_t OFF_HM   = OFF_HC   + sizeof(unsigned) * kNB1;     // hist masses
constexpr size_t OFF_SCC  = OFF_HM   + sizeof(float)    * kNB1;     // scan counts
constexpr size_t OFF_SCP  = OFF_SCC  + sizeof(unsigned) * kNT;      // scan masses
constexpr size_t OFF_TOK  = OFF_SCP  + sizeof(float)    * kNT;      // token ids
constexpr size_t OFF_PEN  = OFF_TOK  + sizeof(int)      * kHistMax; // token penalties
constexpr size_t OFF_MASK = OFF_PEN  + sizeof(float)    * kHistMax; // vocab bitmask
constexpr size_t OFF_SCAL = (OFF_MASK + sizeof(unsigned) * kMaskW + 15) & ~size_t(15);
constexpr size_t kSmemBytes = OFF_SCAL + ((sizeof(Scal) + 15) & ~size_t(15));
// ~244 KB total, fits the 320 KB CDNA5 WGP LDS.

// Monotone float->uint key: x > y  <=>  key(x) > key(y)
__device__ __forceinline__ unsigned keyOf(float x) {
  unsigned u = __float_as_uint(x);
  return (u & 0x80000000u) ? ~u : (u | 0x80000000u);
}

// Cooperative descending scan over NB histogram bins.
// Finds the first bin (descending key order) where including the whole bin
// would violate top-k (C+cnt > k) or nucleus (P+mass > pS).
__device__ void scan_cut(const unsigned* hc, const float* hm, int NB,
                         unsigned startC, float startP, unsigned k,
                         bool computePS, float top_p,
                         unsigned* scanC, float* scanP, Scal* sc) {
  const int tid = threadIdx.x;
  const int per = NB / kNT;  // 16 (level 1/2) or 1 (level 3)

  unsigned cC = 0; float cP = 0.0f;
  for (int j = 0; j < per; ++j) {
    int b = NB - 1 - (tid * per + j);
    cC += hc[b];
    cP += hm[b];
  }
  scanC[tid] = cC; scanP[tid] = cP;
  __syncthreads();

  // Hillis-Steele inclusive scan over 256 partials
  for (int off = 1; off < kNT; off <<= 1) {
    unsigned aC = 0; float aP = 0.0f;
    if (tid >= off) { aC = scanC[tid - off]; aP = scanP[tid - off]; }
    __syncthreads();
    if (tid >= off) { scanC[tid] += aC; scanP[tid] += aP; }
    __syncthreads();
  }
  const unsigned inclC = scanC[tid];
  const float    inclP = scanP[tid];
  const unsigned exC = inclC - cC;
  const float    exP = inclP - cP;

  if (tid == kNT - 1) {
    sc->totalC = startC + inclC;
    sc->totalP = startP + inclP;
    if (computePS) sc->pS = top_p * (startP + inclP);  // p * S_total
    sc->found = 0u;
    sc->minT  = 0x7FFFFFFF;
  }
  __syncthreads();

  const float pS = sc->pS;
  unsigned C = startC + exC;
  float    P = startP + exP;
  int hit = -1;
  unsigned hb = 0, hC = 0, hcnt = 0; float hP = 0.0f, hmass = 0.0f;
  for (int j = 0; j < per; ++j) {
    int b = NB - 1 - (tid * per + j);
    unsigned cb = hc[b];
    if (cb) {
      float mb = hm[b];
      if ((C + cb > k) || (P + mb > pS)) {
        hit = j; hb = (unsigned)b; hC = C; hP = P; hcnt = cb; hmass = mb;
        break;
      }
      C += cb; P += mb;
    }
  }
  if (hit >= 0) atomicMin(&sc->minT, tid);
  __syncthreads();
  if (hit >= 0 && sc->minT == tid) {
    sc->found = 1u; sc->bin = hb;
    sc->Cb = hC; sc->Pb = hP; sc->cntb = hcnt; sc->massb = hmass;
  }
  __syncthreads();
}

}  // namespace

__global__ __launch_bounds__(kNT) void sampler_kernel(
    const float* __restrict__ logits,
    const float* __restrict__ presence,
    const float* __restrict__ frequency,
    const float* __restrict__ temps,
    const float* __restrict__ top_ps,
    const int*   __restrict__ out_tokens,
    const int*   __restrict__ top_ks,
    float*       __restrict__ probs,
    int hist) {
  extern __shared__ char smem[];
  float*    z     = (float*)(smem + OFF_Z);
  unsigned* hc    = (unsigned*)(smem + OFF_HC);
  float*    hm    = (float*)(smem + OFF_HM);
  unsigned* scanC = (unsigned*)(smem + OFF_SCC);
  float*    scanP = (float*)(smem + OFF_SCP);
  int*      tok   = (int*)(smem + OFF_TOK);
  float*    pen   = (float*)(smem + OFF_PEN);
  unsigned* mask  = (unsigned*)(smem + OFF_MASK);
  Scal*     sc    = (Scal*)(smem + OFF_SCAL);

  const int row = blockIdx.x;
  const int tid = threadIdx.x;

  const float p_pen = presence[row];
  const float f_pen = frequency[row];
  const float invT  = 1.0f / temps[row];
  const float top_p = top_ps[row];
  const unsigned k  = (unsigned)top_ks[row];
  const float* base = logits + (size_t)row * kV;

  // ---- Phase 0: kick off the row DMA (global -> LDS) FIRST, so it overlaps
  //      the token/penalty setup below. One TDM descriptor moves all 201 KB.
#if HAVE_TDM
  if (tid < 32) {  // one wave issues the tensor DMA (EXEC is ignored by TDM)
    typedef unsigned int v4u __attribute__((ext_vector_type(4)));
    typedef int v8i __attribute__((ext_vector_type(8)));
    typedef int v4i __attribute__((ext_vector_type(4)));
    const unsigned long long ga = (unsigned long long)(uintptr_t)base;
    const unsigned lds_addr =
        (unsigned)__builtin_amdgcn_groupstaticsize() + (unsigned)OFF_Z;
    // D# group0: count=1 | lds_addr | global_addr[56:0] | type=2("image")
    v4u g0;
    g0.x = 1u;                                        // count=1, user mode
    g0.y = lds_addr;                                  // LDS byte address
    g0.z = (unsigned)(ga & 0xFFFFFFFFull);            // global_addr[31:0]
    g0.w = (unsigned)((ga >> 32) & 0x01FFFFFFull)     // global_addr[56:32]
         | (2u << 30);                                // descriptor type = 2
    // D# group1: data_size=4B; tensor_dim0=V; tensor_dim1=1; tile=Vx1;
    //            tensor_dim0_stride=V; no padding/iterate/multicast.
    const unsigned td0 = (unsigned)kV;
    v8i g1;
    g1[0] = (int)(2u << 16);                          // data_size code 2 = 4B
    g1[1] = (int)((td0 & 0xFFFFu) << 16);             // tensor_dim0[15:0]
    g1[2] = (int)((td0 >> 16) | (1u << 16));          // tensor_dim0[31:16] | tensor_dim1.lo=1
    g1[3] = (int)((td0 & 0xFFFFu) << 16);             // tile_dim0 = V (fits 16 bits)
    g1[4] = 1;                                        // tile_dim1 = 1
    g1[5] = (int)td0;                                 // tensor_dim0_stride[31:0]
    g1[6] = 0;                                        // stride0 hi / stride1 lo
    g1[7] = 0;
    v4i g2 = {0, 0, 0, 0};                            // unused (<=2D tensor)
    v4i g3 = {0, 0, 0, 0};
    v8i g4 = {0, 0, 0, 0, 0, 0, 0, 0};                // unused trailing group
    __builtin_amdgcn_tensor_load_to_lds(g0, g1, g2, g3, g4, 0);
  }
#elif HAVE_ASYNC_LDS
  for (int i = tid; i < kV; i += kNT) {
    __builtin_amdgcn_global_load_async_to_lds_b32(
        (__attribute__((address_space(1))) int*)(base + i),
        (__attribute__((address_space(3))) int*)(z + i),
        0, 0);
  }
#else
  for (int i = tid; i < kV; i += kNT) z[i] = base[i];
#endif

  // ---- Phase 1: token table, per-token penalty, vocab bitmask ------------
  //      (runs while the DMA streams the logits into LDS)
  for (int j = tid; j < hist; j += kNT) tok[j] = out_tokens[row * hist + j];
  for (int i = tid; i < kMaskW; i += kNT) mask[i] = 0u;
  __syncthreads();
  for (int j = tid; j < hist; j += kNT) {
    int t = tok[j];
    int c = 0;
    for (int q = 0; q < hist; ++q) c += (tok[q] == t) ? 1 : 0;
    pen[j] = f_pen * (float)c + p_pen;           // total subtractive penalty
    atomicOr(&mask[(unsigned)t >> 5], 1u << ((unsigned)t & 31u));
  }

  // ---- wait for the row to land in LDS -----------------------------------
#if HAVE_TDM
  if (tid < 32) __builtin_amdgcn_s_wait_tensorcnt(0);
#elif HAVE_ASYNC_LDS
#if HAVE_WAIT_ASYNC
  __builtin_amdgcn_s_wait_asynccnt(0);
#else
  asm volatile("s_wait_asynccnt 0" ::: "memory");
#endif
#endif
  __syncthreads();

  // ---- Phase 2: apply penalties + temperature in LDS, block-max ----------
  float lm = -3.4e38f;
  for (int i = tid; i < kV; i += kNT) {
    float x = z[i];
    if ((mask[i >> 5] >> (i & 31)) & 1u) {
      for (int j = 0; j < hist; ++j) {
        if (tok[j] == i) { x -= pen[j]; break; }
      }
    }
    x *= invT;
    z[i] = x;
    lm = fmaxf(lm, x);
  }
  scanP[tid] = lm;
  __syncthreads();
  for (int off = kNT / 2; off > 0; off >>= 1) {
    if (tid < off) scanP[tid] = fmaxf(scanP[tid], scanP[tid + off]);
    __syncthreads();
  }
  if (tid == 0) sc->M = scanP[0];
  __syncthreads();
  const float M = sc->M;

  // ---- Phase 3: level-1 radix histogram (key[31:20]) ---------------------
  for (int i = tid; i < kNB1; i += kNT) { hc[i] = 0u; hm[i] = 0.0f; }
  __syncthreads();
  for (int i = tid; i < kV; i += kNT) {
    float x = z[i];
    unsigned b = keyOf(x) >> 20;
    atomicAdd(&hc[b], 1u);
    atomicAdd(&hm[b], __expf(x - M));
  }
  __syncthreads();
  scan_cut(hc, hm, kNB1, 0u, 0.0f, k, /*computePS=*/true, top_p, scanC, scanP, sc);

  unsigned T;     // keep element iff keyOf(z) >= T
  float Skept;    // mass of kept set
  if (sc->found == 0u) {
    T = 0u; Skept = sc->totalP;                  // everything kept
  } else {
    const unsigned b1 = sc->bin;
    const unsigned C1 = sc->Cb;  const float P1 = sc->Pb;
    const float    m1 = sc->massb;

    // ---- level-2: key[19:8] within bin b1 --------------------------------
    for (int i = tid; i < kNB1; i += kNT) { hc[i] = 0u; hm[i] = 0.0f; }
    __syncthreads();
    for (int i = tid; i < kV; i += kNT) {
      float x = z[i];
      unsigned key = keyOf(x);
      if ((key >> 20) == b1) {
        unsigned b = (key >> 8) & 0xFFFu;
        atomicAdd(&hc[b], 1u);
        atomicAdd(&hm[b], __expf(x - M));
      }
    }
    __syncthreads();
    scan_cut(hc, hm, kNB1, C1, P1, k, false, top_p, scanC, scanP, sc);

    if (sc->found == 0u) {
      T = b1 << 20; Skept = P1 + m1;             // keep whole bin (fallback)
    } else {
      const unsigned b2  = sc->bin;
      const unsigned C2  = sc->Cb;  const float P2 = sc->Pb;
      const float    m2  = sc->massb;
      const unsigned pre = (b1 << 12) | b2;

      // ---- level-3: key[7:0] within sub-bin ------------------------------
      for (int i = tid; i < 256; i += kNT) { hc[i] = 0u; hm[i] = 0.0f; }
      __syncthreads();
      for (int i = tid; i < kV; i += kNT) {
        float x = z[i];
        unsigned key = keyOf(x);
        if ((key >> 8) == pre) {
          unsigned b = key & 0xFFu;
          atomicAdd(&hc[b], 1u);
          atomicAdd(&hm[b], __expf(x - M));
        }
      }
      __syncthreads();
      scan_cut(hc, hm, 256, C2, P2, k, false, top_p, scanC, scanP, sc);

      if (sc->found == 0u) {
        T = pre << 8; Skept = P2 + m2;           // keep whole sub-bin
      } else {
        const unsigned b3 = sc->bin;
        const unsigned C3 = sc->Cb;  const float P3 = sc->Pb;
        const float    m3 = sc->massb;
        const unsigned K  = (pre << 8) | b3;     // exact 32-bit key at cutoff
        const bool any = (C3 < k) && (P3 <= sc->pS);
        if (any) { T = K;      Skept = P3 + m3; }  // keep the (single) key
        else     { T = K + 1u; Skept = P3; }       // drop it, keep above
      }
    }
  }
  const float invS = 1.0f / Skept;

  // ---- Phase 4: final probabilities, streamed out non-temporally ---------
  float* outRow = probs + (size_t)row * kV;
  for (int i = tid; i < kV; i += kNT) {
    float x = z[i];
    float v = (keyOf(x) >= T) ? (__expf(x - M) * invS) : 0.0f;
    __builtin_nontemporal_store(v, &outRow[i]);
  }
}

extern "C" void kernel_launch(void* const* d_in, const int* in_sizes, int n_in,
                              void* d_out, int out_size, void* d_ws, size_t ws_size,
                              hipStream_t stream) {
  (void)n_in; (void)out_size; (void)d_ws; (void)ws_size;
  const float* logits = (const float*)d_in[0];
  const float* pres   = (const float*)d_in[1];
  const float* freq   = (const float*)d_in[2];
  const float* temps  = (const float*)d_in[3];
  const float* topp   = (const float*)d_in[4];
  const int*   toks   = (const int*)d_in[5];
  const int*   topk   = (const int*)d_in[6];

  const int Bn   = in_sizes[1];                       // rows
  int hist = (Bn > 0) ? (in_sizes[5] / Bn) : 0;       // history length (200)
  if (hist > kHistMax) hist = kHistMax;

  (void)hipFuncSetAttribute((const void*)sampler_kernel,
                            hipFuncAttributeMaxDynamicSharedMemorySize,
                            (int)kSmemBytes);

  sampler_kernel<<<Bn, kNT, kSmemBytes, stream>>>(
      logits, pres, freq, temps, topp, toks, topk, (float*)d_out, hist);
}